// RecStoreEmbeddingBagCollection_59133109731338
// MI455X (gfx1250) — compile-verified
//
#include <hip/hip_runtime.h>

// ---------------------------------------------------------------------------
// RecStore EmbeddingBagCollection for MI455X (gfx1250)
// Memory-bound random gather (~816 MB/call -> ~35us HBM floor @23.3TB/s).
// Uses the CDNA5 Tensor Data Mover in gather mode (8 rows / descriptor,
// 32-bit indices) with a PIPE-deep software pipeline of LDS buffers,
// one wave32 per bag, TENSORcnt-based synchronization.
// ---------------------------------------------------------------------------

typedef __attribute__((ext_vector_type(4))) unsigned int u32x4;
typedef __attribute__((ext_vector_type(4))) int          i32x4;
typedef __attribute__((ext_vector_type(8))) int          i32x8;

#ifndef __has_builtin
#define __has_builtin(x) 0
#endif

#define TDM_AVAILABLE (__has_builtin(__builtin_amdgcn_tensor_load_to_lds) && \
                       __has_builtin(__builtin_amdgcn_s_wait_tensorcnt))

static constexpr int BAGS = 16384;
static constexpr int OUT_STRIDE = 288; // 64 + 128 + 32 + 64
static constexpr int CH = 8;           // rows per TDM gather (32-bit index limit)

// ---------------------------------------------------------------------------
// TDM gather-mode descriptor issue: gather up to 8 rows of `D` f32 elements
// from table W (rows selected by 32-bit indices) into LDS at lds_byte_off.
// D# layout per CDNA5 ISA (gather mode).
// ---------------------------------------------------------------------------
#if TDM_AVAILABLE
__device__ __forceinline__ void tdm_gather8(const float* W, unsigned lds_byte_off,
                                            const int* bids, int start, int nvalid,
                                            int D, int numRows) {
  int id[8];
#pragma unroll
  for (int k = 0; k < 8; ++k)
    id[k] = (k < nvalid) ? __builtin_amdgcn_readfirstlane(bids[start + k]) : 0;

  const unsigned long long ga = (unsigned long long)(const void*)W;

  u32x4 g0;
  g0[0] = 0xC0000001u;                                        // count=1 | 32b idx | gather_mode
  g0[1] = lds_byte_off;                                       // lds_addr
  g0[2] = (unsigned)(ga & 0xffffffffu);                       // global_addr[31:0]
  g0[3] = ((unsigned)(ga >> 32) & 0x01ffffffu) | (2u << 30);  // global_addr[56:32] | type=2

  i32x8 g1;
  g1[0] = 0x00020000;                                   // workgroup_mask=0, data_size=2 (4B)
  g1[1] = (int)((unsigned)(D & 0xffff) << 16);          // tensor_dim0[15:0] (barrier addr = 0)
  g1[2] = (int)(((unsigned)numRows & 0xffffu) << 16);   // tensor_dim1[15:0] | tensor_dim0 hi = 0
  g1[3] = (int)(((unsigned)D << 16) | ((unsigned)numRows >> 16)); // tile_dim0 | tensor_dim1 hi
  g1[4] = nvalid;                                       // tile_dim1 = #valid indices
  g1[5] = D;                                            // tensor_dim0_stride (elements)
  g1[6] = 0;
  g1[7] = 0;

  i32x4 g2; g2[0] = id[0]; g2[1] = id[1]; g2[2] = id[2]; g2[3] = id[3];
  i32x4 g3; g3[0] = id[4]; g3[1] = id[5]; g3[2] = id[6]; g3[3] = id[7];

#if defined(__clang_major__) && (__clang_major__ >= 23)
  i32x8 gx = {};
  __builtin_amdgcn_tensor_load_to_lds(g0, g1, g2, g3, gx, 0);
#else
  __builtin_amdgcn_tensor_load_to_lds(g0, g1, g2, g3, 0);
#endif
}
#endif // TDM_AVAILABLE

// ---------------------------------------------------------------------------
// Sum-pooling. One wave32 per bag; lane owns D/32 contiguous columns.
// PIPE-deep TDM gather pipeline (prologue / steady / drain) through LDS.
// ---------------------------------------------------------------------------
template <int D, int WAVES, int COLOFF, int PIPE>
__global__ __launch_bounds__(WAVES * 32) void pool_kernel(
    const float* __restrict__ W, const int* __restrict__ ids,
    const int* __restrict__ lens, const int* __restrict__ offs,
    float* __restrict__ out, int numRows) {
  constexpr int COLS = D / 32;
  __shared__ __align__(16) float buf[WAVES][PIPE][CH * D];

  const int w    = threadIdx.x >> 5;
  const int lane = threadIdx.x & 31;
  const int bag  = blockIdx.x * WAVES + w;

  const int n   = __builtin_amdgcn_readfirstlane(lens[bag]);
  const int off = __builtin_amdgcn_readfirstlane(offs[bag]);
  const int* bids = ids + off;

  float acc[COLS];
#pragma unroll
  for (int k = 0; k < COLS; ++k) acc[k] = 0.0f;

#if TDM_AVAILABLE
  const int nch = (n + CH - 1) / CH;
  if (n > 0) {
    // Prologue: fill the pipeline.
    const int pre = (nch < PIPE) ? nch : PIPE;
    for (int i = 0; i < pre; ++i) {
      const int s0 = i * CH;
      const int nv = (n - s0 < CH) ? (n - s0) : CH;
      tdm_gather8(W, (unsigned)(size_t)(void*)&buf[w][i % PIPE][0], bids, s0, nv,
                  D, numRows);
    }
    int c = 0;
    // Steady state: chunks c .. c+PIPE-1 are in flight.
    for (; c + PIPE < nch; ++c) {
      __builtin_amdgcn_s_wait_tensorcnt(PIPE - 1); // oldest chunk (c) landed
      const int si = c + PIPE;
      const int nvi = (n - si * CH < CH) ? (n - si * CH) : CH;
      tdm_gather8(W, (unsigned)(size_t)(void*)&buf[w][si % PIPE][0], bids,
                  si * CH, nvi, D, numRows);
      const float* lb = &buf[w][c % PIPE][0];
#pragma unroll
      for (int r = 0; r < CH; ++r)
#pragma unroll
        for (int k = 0; k < COLS; ++k)
          acc[k] += lb[r * D + lane * COLS + k];
    }
    // Drain: everything issued; wait once, then consume remaining buffers.
    __builtin_amdgcn_s_wait_tensorcnt(0);
    for (; c < nch; ++c) {
      const int nv = (n - c * CH < CH) ? (n - c * CH) : CH;
      const float* lb = &buf[w][c % PIPE][0];
      for (int r = 0; r < nv; ++r)
#pragma unroll
        for (int k = 0; k < COLS; ++k)
          acc[k] += lb[r * D + lane * COLS + k];
    }
  }
#else
  // Fallback: scalar id + wide vector row loads, accumulate in VGPRs.
  for (int j = 0; j < n; ++j) {
    const int id = __builtin_amdgcn_readfirstlane(bids[j]);
    const float* row = W + (size_t)id * D + lane * COLS;
#pragma unroll
    for (int k = 0; k < COLS; ++k) acc[k] += row[k];
  }
#endif

  float* o = out + (size_t)bag * OUT_STRIDE + COLOFF + lane * COLS;
#pragma unroll
  for (int k = 0; k < COLS; ++k) o[k] = acc[k];
}

// Explicit instantiations first so the pool kernels lead the asm file.
template __global__ void pool_kernel<64, 4, 0, 4>(
    const float*, const int*, const int*, const int*, float*, int);
template __global__ void pool_kernel<128, 4, 64, 2>(
    const float*, const int*, const int*, const int*, float*, int);
template __global__ void pool_kernel<32, 8, 192, 4>(
    const float*, const int*, const int*, const int*, float*, int);
template __global__ void pool_kernel<64, 4, 224, 4>(
    const float*, const int*, const int*, const int*, float*, int);

// ---------------------------------------------------------------------------
// Exclusive prefix sum of lens[16384] per table (one block / table).
// ---------------------------------------------------------------------------
__global__ __launch_bounds__(1024) void exscan_all(
    const int* __restrict__ l0, const int* __restrict__ l1,
    const int* __restrict__ l2, const int* __restrict__ l3,
    int* __restrict__ offs) {
  __shared__ int s[1024];
  const int* lp[4] = {l0, l1, l2, l3};
  const int* lens = lp[blockIdx.x];
  int* o = offs + blockIdx.x * BAGS;

  const int t = threadIdx.x;
  const int base = t * 16; // 16384 / 1024
  int local[16];
  int tot = 0;
#pragma unroll
  for (int k = 0; k < 16; ++k) { local[k] = lens[base + k]; tot += local[k]; }
  s[t] = tot;
  __syncthreads();
  for (int d = 1; d < 1024; d <<= 1) { // Hillis-Steele inclusive scan
    int v = (t >= d) ? s[t - d] : 0;
    __syncthreads();
    s[t] += v;
    __syncthreads();
  }
  int run = s[t] - tot; // exclusive prefix of this thread's chunk
#pragma unroll
  for (int k = 0; k < 16; ++k) { o[base + k] = run; run += local[k]; }
}

// ---------------------------------------------------------------------------
// Launch: inputs are W0,ids0,lens0, W1,ids1,lens1, W2,ids2,lens2, W3,ids3,lens3
// ---------------------------------------------------------------------------
extern "C" void kernel_launch(void* const* d_in, const int* in_sizes, int n_in,
                              void* d_out, int out_size, void* d_ws, size_t ws_size,
                              hipStream_t stream) {
  const int dims[4] = {64, 128, 32, 64};
  const float* W[4];
  const int* ids[4];
  const int* lens[4];
  int numRows[4];
  for (int t = 0; t < 4; ++t) {
    W[t]    = (const float*)d_in[3 * t + 0];
    ids[t]  = (const int*)d_in[3 * t + 1];
    lens[t] = (const int*)d_in[3 * t + 2];
    numRows[t] = in_sizes[3 * t + 0] / dims[t];
  }

  int* offs = (int*)d_ws; // 4 * 16384 int32 = 256 KiB of scratch

  exscan_all<<<4, 1024, 0, stream>>>(lens[0], lens[1], lens[2], lens[3], offs);

  float* out = (float*)d_out;
  // Column offsets in the concatenated output: 0, 64, 192, 224 (stride 288).
  pool_kernel<64, 4, 0, 4>   <<<BAGS / 4, 128, 0, stream>>>(W[0], ids[0], lens[0], offs + 0 * BAGS, out, numRows[0]);
  pool_kernel<128, 4, 64, 2> <<<BAGS / 4, 128, 0, stream>>>(W[1], ids[1], lens[1], offs + 1 * BAGS, out, numRows[1]);
  pool_kernel<32, 8, 192, 4> <<<BAGS / 8, 256, 0, stream>>>(W[2], ids[2], lens[2], offs + 2 * BAGS, out, numRows[2]);
  pool_kernel<64, 4, 224, 4> <<<BAGS / 4, 128, 0, stream>>>(W[3], ids[3], lens[3], offs + 3 * BAGS, out, numRows[3]);
}